// SingleValueAgent_67095979098188
// MI455X (gfx1250) — compile-verified
//
#include <hip/hip_runtime.h>
#include <cmath>

typedef __attribute__((ext_vector_type(2))) float v2f;
typedef __attribute__((ext_vector_type(4))) float v4f;
typedef __attribute__((ext_vector_type(8))) float v8f;

#define N_TRIALS 2048
#define CHUNK 16

__device__ __forceinline__ float powi_f(float b, int e) {
    float r = 1.0f;
    for (int i = 0; i < e; ++i) r *= b;
    return r;
}

// One wave (32 threads) per block; each wave owns 16 sessions and scans all
// 2048 trials in 16-trial chunks. Each chunk is a 16x16 triangular matmul on
// the f32 WMMA unit: V = L @ X + p (x) carry.
__global__ __launch_bounds__(32) void sva_wmma_scan(
    const float* __restrict__ in,     // (n_sess, N_TRIALS, 2) f32
    const float* __restrict__ a_raw,  // (1,) f32, pre-sigmoid alpha
    float* __restrict__ out,          // [v | diff | delta], each n_elem f32
    long long n_elem)                 // n_sess * N_TRIALS
{
    // v-tile transpose buffer: 16 rows (trials) x 16 cols (sessions), pad 17.
    // Row 16 (offset 272) stashes the chunk-entry carry per session.
    __shared__ float sm[17 * 16 + 16];

    const int  lane = threadIdx.x;      // 0..31
    const bool hi   = lane >= 16;       // high lane half
    const int  s    = lane & 15;        // session column / A-matrix row M
    const int  s0   = blockIdx.x * 16;  // first session of this wave's group

    // alpha = sigmoid(a_raw)
    const float ar        = a_raw[0];
    const float alpha     = 1.0f / (1.0f + __expf(-ar));
    const float om        = 1.0f - alpha;      // (1 - alpha)
    const float inv_alpha = 1.0f / alpha;

    // A fragments: constant lower-triangular L, L[m][k] = alpha*om^(m-k), k<=m.
    // 16x4 f32 A layout: lanes 0-15 hold M=lane; VGPR0 = K (low half: k0+0,
    // high half: k0+2), VGPR1 = k0+1 / k0+3.
    v2f A[4];
#pragma unroll
    for (int i = 0; i < 4; ++i) {
        int kx = 4 * i + (hi ? 2 : 0);
        int ky = kx + 1;
        A[i].x = (kx <= s) ? alpha * powi_f(om, s - kx) : 0.0f;
        A[i].y = (ky <= s) ? alpha * powi_f(om, s - ky) : 0.0f;
    }

    // Carry-propagation coefficients for the C accumulator:
    // C/D layout: VGPR j = row t, t = j (lanes 0-15) or j+8 (lanes 16-31).
    // v_t gets (1-alpha)^(t+1) * carry.
    v8f pv;
#pragma unroll
    for (int j = 0; j < 8; ++j) {
        int t = j + (hi ? 8 : 0);
        pv[j] = powi_f(om, t + 1);
    }

    // Write-pass mapping: each lane owns 4 consecutive trials of one session
    // so every output store is a 16B global_store_b128 (4-lane groups cover
    // 64B contiguous per session row).
    const size_t srow = (size_t)(s0 + s) * (2 * N_TRIALS);
    float carry = 0.0f;  // v_{-1} = 0

    for (int cb = 0; cb < N_TRIALS; cb += CHUNK) {
        // Stash chunk-entry carry for the t==0 diff in the write pass.
        if (!hi) sm[272 + s] = carry;

        // B fragments: X[k][s] = choice*outcome at trial cb+k, session s.
        // 4x16 f32 B layout mirrors A: VGPR0 = K (k0+0 low / k0+2 high),
        // VGPR1 = k0+1 / k0+3. Two consecutive trials => one 16B load.
        v2f B[4];
#pragma unroll
        for (int i = 0; i < 4; ++i) {
            int t0 = cb + 4 * i + (hi ? 2 : 0);
            v4f f  = __builtin_nontemporal_load(
                         reinterpret_cast<const v4f*>(in + srow + (size_t)t0 * 2));
            B[i].x = f.x * f.y;   // trial t0
            B[i].y = f.z * f.w;   // trial t0 + 1
        }

        // C init = p (x) carry, then D = L @ X + C via 4 chained K=4 WMMAs.
        v8f c;
#pragma unroll
        for (int j = 0; j < 8; ++j) c[j] = pv[j] * carry;

        c = __builtin_amdgcn_wmma_f32_16x16x4_f32(false, A[0], false, B[0],
                                                  (short)0, c, false, false);
        c = __builtin_amdgcn_wmma_f32_16x16x4_f32(false, A[1], false, B[1],
                                                  (short)0, c, false, false);
        c = __builtin_amdgcn_wmma_f32_16x16x4_f32(false, A[2], false, B[2],
                                                  (short)0, c, false, false);
        c = __builtin_amdgcn_wmma_f32_16x16x4_f32(false, A[3], false, B[3],
                                                  (short)0, c, false, false);

        // New carry = v[15][s], held in VGPR7 of high-half lane 16+s.
        carry = __shfl(c[7], 16 + s, 32);

        // Drop the v-tile into LDS (trial-major, padded rows).
#pragma unroll
        for (int j = 0; j < 8; ++j)
            sm[(j + (hi ? 8 : 0)) * 17 + s] = c[j];

        __syncthreads();  // single-wave WG: free barrier, acts as LDS fence

        // Transposed write pass: 2 iterations x 32 lanes cover the 16x16
        // tile as (session, trial-quad) pairs. Lane reads its 4 v values +
        // the predecessor value down an LDS column, derives delta/diff, and
        // emits one non-temporal b128 store per output array.
#pragma unroll
        for (int it = 0; it < 2; ++it) {
            int p  = it * 32 + lane;
            int s2 = p >> 2;        // session 0..15
            int q  = p & 3;         // trial quad 0..3
            int tb = q * 4;         // first trial of the quad

            float v0 = sm[(tb + 0) * 17 + s2];
            float v1 = sm[(tb + 1) * 17 + s2];
            float v2 = sm[(tb + 2) * 17 + s2];
            float v3 = sm[(tb + 3) * 17 + s2];
            float vm = (q == 0) ? sm[272 + s2] : sm[(tb - 1) * 17 + s2];

            v4f vv = {v0, v1, v2, v3};
            v4f dd = {v0 - vm, v1 - v0, v2 - v1, v3 - v2};  // delta
            v4f ff = dd * inv_alpha;                         // diff

            size_t idx = (size_t)(s0 + s2) * N_TRIALS + (size_t)(cb + tb);
            __builtin_nontemporal_store(vv, reinterpret_cast<v4f*>(out + idx));
            __builtin_nontemporal_store(ff, reinterpret_cast<v4f*>(out + (size_t)n_elem + idx));
            __builtin_nontemporal_store(dd, reinterpret_cast<v4f*>(out + 2 * (size_t)n_elem + idx));
        }

        __syncthreads();  // fence before next chunk reuses the LDS tile
    }
}

extern "C" void kernel_launch(void* const* d_in, const int* in_sizes, int n_in,
                              void* d_out, int out_size, void* d_ws, size_t ws_size,
                              hipStream_t stream) {
    const float* in    = (const float*)d_in[0];   // (8192, 2048, 2) f32
    const float* a_raw = (const float*)d_in[1];   // (1,) f32
    float*       out   = (float*)d_out;           // 3 * n_elem f32

    const long long n_elem = (long long)in_sizes[0] / 2;      // 8192*2048
    const int       n_sess = (int)(n_elem / N_TRIALS);        // 8192

    dim3 grid(n_sess / 16);  // 512 waves, 16 sessions each
    dim3 block(32);
    hipLaunchKernelGGL(sva_wmma_scan, grid, block, 0, stream,
                       in, a_raw, out, n_elem);
}